// Basicgate_patch_iv_multivoxel_66726611910700
// MI455X (gfx1250) — compile-verified
//
#include <hip/hip_runtime.h>
#include <math.h>

#define H_    256
#define W_    704
#define HW    (H_ * W_)          // 180224
#define C_IMG 256
#define COUT  131
#define MPAD  144                // 9 * 16
#define KPAD  132                // 33 * 4

typedef __attribute__((ext_vector_type(2))) float v2f;
typedef __attribute__((ext_vector_type(8))) float v8f;

__device__ __forceinline__ v8f wmma_f32(v2f a, v2f b, v8f c) {
  // 8 args: (neg_a, A, neg_b, B, c_mod, C, reuse_a, reuse_b)
  return __builtin_amdgcn_wmma_f32_16x16x4_f32(false, a, false, b, (short)0, c,
                                               false, false);
}

// ---------------------------------------------------------------- init S with rd0_b + rd1_b
__global__ void k_init_s(float* __restrict__ S, const float* __restrict__ b0,
                         const float* __restrict__ b1) {
  int c = blockIdx.y;
  int p = (blockIdx.x * blockDim.x + threadIdx.x) * 4;
  float v = b0[c] + b1[c];
  float4 q = make_float4(v, v, v, v);
  *(float4*)(S + (size_t)c * HW + p) = q;
}

// ---------------------------------------------------------------- pad W2 (131x131 -> 144x132) + b2
__global__ void k_pad_w2(const float* __restrict__ w2, const float* __restrict__ b2,
                         float* __restrict__ w2p, float* __restrict__ b2p) {
  for (int i = blockIdx.x * blockDim.x + threadIdx.x; i < MPAD * KPAD;
       i += gridDim.x * blockDim.x) {
    int r = i / KPAD;
    int c = i - r * KPAD;
    w2p[i] = (r < COUT && c < COUT) ? w2[r * COUT + c] : 0.f;
  }
  int t = blockIdx.x * blockDim.x + threadIdx.x;
  if (t < MPAD) b2p[t] = (t < COUT) ? b2[t] : 0.f;
}

// ---------------------------------------------------------------- scatter + per-point matvec
template <int CF>
__global__ void k_scatter_conv(const float* __restrict__ feat,
                               const float* __restrict__ coord,
                               const int* __restrict__ grid,
                               const float* __restrict__ w, int npts,
                               float* __restrict__ S) {
  int idx = blockIdx.x * blockDim.x + threadIdx.x;
  int total = npts * COUT;
  if (idx >= total) return;
  int pt = idx / COUT;
  int co = idx - pt * COUT;
  int col = grid[pt * 2 + 0], row = grid[pt * 2 + 1];
  if ((unsigned)row >= H_ || (unsigned)col >= W_) return;
  const float* wr = w + co * (CF + 3);
  const float* f = feat + (size_t)pt * CF;
  float acc = 0.f;
#pragma unroll 8
  for (int k = 0; k < CF; ++k) acc += wr[k] * f[k];
  const float* cd = coord + (size_t)pt * 3;
  acc += wr[CF + 0] * cd[0] + wr[CF + 1] * cd[1] + wr[CF + 2] * cd[2];
  atomicAdd(S + (size_t)co * HW + row * W_ + col, acc);
}

__global__ void k_scatter_raw(const float* __restrict__ feat,
                              const float* __restrict__ coord,
                              const int* __restrict__ grid, int npts,
                              float* __restrict__ S) {
  int idx = blockIdx.x * blockDim.x + threadIdx.x;
  int total = npts * COUT;
  if (idx >= total) return;
  int pt = idx / COUT;
  int c = idx - pt * COUT;
  int col = grid[pt * 2 + 0], row = grid[pt * 2 + 1];
  if ((unsigned)row >= H_ || (unsigned)col >= W_) return;
  float v = (c < 128) ? feat[(size_t)pt * 128 + c] : coord[pt * 3 + (c - 128)];
  atomicAdd(S + (size_t)c * HW + row * W_ + col, v);
}

// ---------------------------------------------------------------- gated = rd3 . img + b3
__global__ void k_gated(const float* __restrict__ img, const float* __restrict__ w3,
                        const float* __restrict__ b3, float* __restrict__ gated) {
  int p4 = blockIdx.x * blockDim.x + threadIdx.x;  // HW/4 threads
  const float4* img4 = (const float4*)img;
  float b = b3[0];
  float4 acc = make_float4(b, b, b, b);
  const int stride = HW / 4;
  for (int c = 0; c < C_IMG; ++c) {
    float4 v = img4[(size_t)c * stride + p4];
    float wv = w3[c];
    acc.x += v.x * wv; acc.y += v.y * wv;
    acc.z += v.z * wv; acc.w += v.w * wv;
  }
  ((float4*)gated)[p4] = acc;
}

// ---------------------------------------------------------------- rd2 GEMM via V_WMMA_F32_16X16X4_F32
#define PIXB 64   // pixels per block
#define SST  80   // LDS row stride (floats): rows k and k+2 land in disjoint banks
__global__ __launch_bounds__(128) void k_gemm_rd2(float* __restrict__ S,  // in/out
                                                  const float* __restrict__ w2p,
                                                  const float* __restrict__ b2p,
                                                  const float* __restrict__ gated) {
  __shared__ float sS[KPAD * SST];
  const int tid = threadIdx.x;
  const int p0 = blockIdx.x * PIXB;
  // stage S[0:131][p0:p0+64] into LDS
  for (int i = tid; i < COUT * PIXB; i += 128) {
    int r = i >> 6;
    int c = i & 63;
    sS[r * SST + c] = S[(size_t)r * HW + p0 + c];
  }
  if (tid < PIXB) sS[131 * SST + tid] = 0.f;  // zero-pad K row 131
  __syncthreads();

  const int lane = tid & 31;
  const int wv = tid >> 5;     // 4 waves, 16 pixels each
  const int lh = lane >> 4;    // lane half
  const int ln = lane & 15;
  const int pb = wv * 16;

  const int n = p0 + pb + ln;
  const float g = gated[n];

  for (int mt = 0; mt < 9; ++mt) {
    // A: M = lane%16, lane-half selects K pair {k0,k0+1} vs {k0+2,k0+3}
    const float* wrow = w2p + (mt * 16 + ln) * KPAD + 2 * lh;
    v8f acc = {};
#pragma unroll 3
    for (int kb = 0; kb < 33; ++kb) {
      v2f a = *(const v2f*)(wrow + kb * 4);          // aligned 8B -> global_load_b64
      int k0 = kb * 4 + 2 * lh;
      v2f b; b.x = sS[k0 * SST + pb + ln];
             b.y = sS[(k0 + 1) * SST + pb + ln];
      acc = wmma_f32(a, b, acc);
    }
    // C/D layout: row m = mt*16 + r + 8*(lane/16); bias rows contiguous per lane
    const int m0 = mt * 16 + 8 * lh;
    float4 bl = *(const float4*)(b2p + m0);
    float4 bh = *(const float4*)(b2p + m0 + 4);
    float* dst = S + (size_t)m0 * HW + n;
    if (mt < 8) {  // uniform branch; all 16 rows valid
      dst[0 * HW] = acc[0] + bl.x + g;
      dst[1 * HW] = acc[1] + bl.y + g;
      dst[2 * HW] = acc[2] + bl.z + g;
      dst[3 * HW] = acc[3] + bl.w + g;
      dst[4 * HW] = acc[4] + bh.x + g;
      dst[5 * HW] = acc[5] + bh.y + g;
      dst[6 * HW] = acc[6] + bh.z + g;
      dst[7 * HW] = acc[7] + bh.w + g;
    } else {       // tail tile: rows 128..143, only m < 131 stored
      float bias[8] = {bl.x, bl.y, bl.z, bl.w, bh.x, bh.y, bh.z, bh.w};
#pragma unroll
      for (int r = 0; r < 8; ++r)
        if (m0 + r < COUT) dst[r * HW] = acc[r] + bias[r] + g;
    }
  }
}

// ---------------------------------------------------------------- 3x3 spatial conv -> sigmoid
__global__ __launch_bounds__(256) void k_spatial(const float* __restrict__ F,
                                                 const float* __restrict__ sw,
                                                 const float* __restrict__ sb,
                                                 float* __restrict__ att) {
  __shared__ float tile[18 * 20];
  const int tx = threadIdx.x & 15, ty = threadIdx.x >> 4;
  const int bx = blockIdx.x * 16, by = blockIdx.y * 16;
  float acc = sb[0];
  for (int c = 0; c < COUT; ++c) {
    const float* Fc = F + (size_t)c * HW;
    for (int i = threadIdx.x; i < 18 * 18; i += 256) {
      int hy = i / 18, hx = i - hy * 18;
      int gy = by + hy - 1, gx = bx + hx - 1;
      float v = 0.f;
      if ((unsigned)gy < H_ && (unsigned)gx < W_) v = Fc[gy * W_ + gx];
      tile[hy * 20 + hx] = v;
    }
    __syncthreads();
    const float* wc = sw + c * 9;
    const float* t = tile + ty * 20 + tx;
    acc += wc[0] * t[0]  + wc[1] * t[1]  + wc[2] * t[2]
         + wc[3] * t[20] + wc[4] * t[21] + wc[5] * t[22]
         + wc[6] * t[40] + wc[7] * t[41] + wc[8] * t[42];
    __syncthreads();
  }
  float a = 1.f / (1.f + expf(-acc));
  att[(by + ty) * W_ + bx + tx] = a;
}

// ---------------------------------------------------------------- out = img * att (broadcast)
__global__ void k_mul(const float* __restrict__ img, const float* __restrict__ att,
                      float* __restrict__ out) {
  size_t i = (size_t)blockIdx.x * blockDim.x + threadIdx.x;  // over C_IMG*HW/4
  size_t p4 = i % (HW / 4);
  float4 v = ((const float4*)img)[i];
  float4 a = ((const float4*)att)[p4];
  ((float4*)out)[i] = make_float4(v.x * a.x, v.y * a.y, v.z * a.z, v.w * a.w);
}

extern "C" void kernel_launch(void* const* d_in, const int* in_sizes, int n_in,
                              void* d_out, int out_size, void* d_ws, size_t ws_size,
                              hipStream_t stream) {
  const float* img  = (const float*)d_in[0];
  const float* vf0  = (const float*)d_in[1];
  const float* vc0  = (const float*)d_in[2];
  const int*   g0   = (const int*)d_in[3];
  const float* vf1  = (const float*)d_in[4];
  const float* vc1  = (const float*)d_in[5];
  const int*   g1   = (const int*)d_in[6];
  const float* vf2  = (const float*)d_in[7];
  const float* vc2  = (const float*)d_in[8];
  const int*   g2   = (const int*)d_in[9];
  const float* rd0w = (const float*)d_in[10];
  const float* rd0b = (const float*)d_in[11];
  const float* rd1w = (const float*)d_in[12];
  const float* rd1b = (const float*)d_in[13];
  const float* rd2w = (const float*)d_in[14];
  const float* rd2b = (const float*)d_in[15];
  const float* rd3w = (const float*)d_in[16];
  const float* rd3b = (const float*)d_in[17];
  const float* spw  = (const float*)d_in[18];
  const float* spb  = (const float*)d_in[19];
  float* out = (float*)d_out;

  float* S     = (float*)d_ws;                // 131*HW floats (becomes fused F in place)
  float* gated = S + (size_t)COUT * HW;       // HW floats
  float* att   = gated + HW;                  // HW floats
  float* w2p   = att + HW;                    // 144*132 floats, zero-padded
  float* b2p   = w2p + MPAD * KPAD;           // 144 floats, zero-padded

  const int N0 = in_sizes[3] / 2, N1 = in_sizes[6] / 2, N2 = in_sizes[9] / 2;

  k_gated<<<dim3(HW / 4 / 256), 256, 0, stream>>>(img, rd3w, rd3b, gated);
  k_init_s<<<dim3(HW / 4 / 256, COUT), 256, 0, stream>>>(S, rd0b, rd1b);
  k_pad_w2<<<80, 256, 0, stream>>>(rd2w, rd2b, w2p, b2p);
  k_scatter_conv<32><<<(N0 * COUT + 255) / 256, 256, 0, stream>>>(vf0, vc0, g0, rd0w, N0, S);
  k_scatter_conv<64><<<(N1 * COUT + 255) / 256, 256, 0, stream>>>(vf1, vc1, g1, rd1w, N1, S);
  k_scatter_raw<<<(N2 * COUT + 255) / 256, 256, 0, stream>>>(vf2, vc2, g2, N2, S);
  k_gemm_rd2<<<HW / PIXB, 128, 0, stream>>>(S, w2p, b2p, gated);
  k_spatial<<<dim3(W_ / 16, H_ / 16), 256, 0, stream>>>(S, spw, spb, att);
  k_mul<<<(C_IMG * HW / 4) / 256, 256, 0, stream>>>(img, att, out);
}